// PatchTokenizer_27960237097639
// MI455X (gfx1250) — compile-verified
//
#include <hip/hip_runtime.h>
#include <hip/hip_bf16.h>

// CDNA5 (gfx1250, wave32) VQ patch tokenizer.
// Pipeline:
//   k_init        : cnorm_half[v] = 0.5*||codebook_v||^2 (ws), zero the loss cell
//   k_patch_emb   : fp32 WMMA GEMM  patch_emb = patches @ W^T + b   (M=16352,K=128,N=128)
//   k_vq_argmin   : fp32 WMMA GEMM  scores = patch_emb @ codebook^T, fused argmax
//                   of (score - cnorm_half) over V=8192, codebook staged via LDS
//                   (32 rows per barrier, dual accumulation chains per tile)
//   k_gather_loss : quantized_st = codebook[token], vq_loss = 2*mean((q-pe)^2)

typedef __attribute__((ext_vector_type(2))) float v2f;
typedef __attribute__((ext_vector_type(8))) float v8f;

#define M_TOT   16352   // B*N = 32*511
#define M_TILES 1022    // M_TOT/16
#define D_MODEL 128
#define VOCAB   8192
#define NPB     511     // patches per batch row
#define LDS_STRIDE 132  // 128 + 4 pad -> conflict-free B-fragment reads

__global__ void k_init(const float* __restrict__ cb, float* __restrict__ cn,
                       float* __restrict__ loss) {
  const int wave = threadIdx.x >> 5, lane = threadIdx.x & 31;
  if (blockIdx.x == 0 && threadIdx.x == 0) *loss = 0.0f;
  const int row = blockIdx.x * 8 + wave;
  if (row < VOCAB) {
    const float* p = cb + row * D_MODEL;
    float s = 0.0f;
#pragma unroll
    for (int i = 0; i < 4; ++i) { float v = p[lane + 32 * i]; s += v * v; }
    for (int mask = 16; mask; mask >>= 1) s += __shfl_xor(s, mask, 32);
    if (lane == 0) cn[row] = 0.5f * s;
  }
}

// GEMM1: one wave per 16-row M-tile, loops over 8 N-tiles of W.
// K=128 accumulated in two independent WMMA chains to break the D->C RAW chain.
__global__ void k_patch_emb(const float* __restrict__ x, const float* __restrict__ W,
                            const float* __restrict__ bias, float* __restrict__ pe) {
  const int wave = threadIdx.x >> 5, lane = threadIdx.x & 31;
  const int mtile = blockIdx.x * 8 + wave;
  if (mtile >= M_TILES) return;                // wave-uniform
  const int nloc = lane & 15, khalf = lane >> 4;

  // A fragment: patch rows are contiguous 128-float windows of x
  const int mg = mtile * 16 + nloc;            // A-row index for this lane
  const int bb = mg / NPB, nn = mg - bb * NPB;
  const float* rowA = x + bb * 32768 + nn * 64;  // L*C=32768, STRIDE*C=64
  v2f af[32];
#pragma unroll
  for (int g = 0; g < 32; ++g)
    af[g] = *(const v2f*)(rowA + 4 * g + 2 * khalf);

  for (int dt = 0; dt < 8; ++dt) {
    v8f acc0 = {}, acc1 = {};
    const float* Wrow = W + (dt * 16 + nloc) * D_MODEL;  // B(k,n) = W[n_g][k]
#pragma unroll
    for (int g = 0; g < 32; g += 2) {
      v2f b0 = *(const v2f*)(Wrow + 4 * g + 2 * khalf);
      v2f b1 = *(const v2f*)(Wrow + 4 * (g + 1) + 2 * khalf);
      acc0 = __builtin_amdgcn_wmma_f32_16x16x4_f32(
          false, af[g], false, b0, (short)0, acc0, false, false);
      acc1 = __builtin_amdgcn_wmma_f32_16x16x4_f32(
          false, af[g + 1], false, b1, (short)0, acc1, false, false);
    }
    const v8f acc = acc0 + acc1;
    const float bv = bias[dt * 16 + nloc];
#pragma unroll
    for (int j = 0; j < 8; ++j) {
      const int mrow = mtile * 16 + j + 8 * khalf;       // C/D layout
      pe[mrow * D_MODEL + dt * 16 + nloc] = acc[j] + bv;
    }
  }
}

// GEMM2 + fused argmax: block = 8 waves = 8 M-tiles; codebook streamed via LDS,
// 32 rows (two 16-wide V-tiles) per barrier; dual WMMA chains per sub-tile.
__global__ void k_vq_argmin(const float* __restrict__ pe, const float* __restrict__ cb,
                            const float* __restrict__ cn, float* __restrict__ tok_f,
                            int* __restrict__ tok_i) {
  __shared__ float lds[32 * LDS_STRIDE];
  const int wave = threadIdx.x >> 5, lane = threadIdx.x & 31;
  const int mtile = blockIdx.x * 8 + wave;
  const bool valid = (mtile < M_TILES);        // wave-uniform
  const int nloc = lane & 15, khalf = lane >> 4;

  v2f af[32];
  if (valid) {
    const float* rowA = pe + (mtile * 16 + nloc) * D_MODEL;
#pragma unroll
    for (int g = 0; g < 32; ++g)
      af[g] = *(const v2f*)(rowA + 4 * g + 2 * khalf);
  }

  float best[8];
  int bidx[8];
#pragma unroll
  for (int j = 0; j < 8; ++j) { best[j] = -3.4e38f; bidx[j] = 0; }

  const int r  = threadIdx.x >> 3;             // staging: code row 0..31
  const int c0 = (threadIdx.x & 7) * 16;       // staging: 16 floats per thread

  for (int st = 0; st < VOCAB / 32; ++st) {    // 256 staged chunks
    __syncthreads();
    {
      const float* src = cb + (st * 32 + r) * D_MODEL + c0;
      float* dst = lds + r * LDS_STRIDE + c0;
      *(float4*)(dst)      = *(const float4*)(src);
      *(float4*)(dst + 4)  = *(const float4*)(src + 4);
      *(float4*)(dst + 8)  = *(const float4*)(src + 8);
      *(float4*)(dst + 12) = *(const float4*)(src + 12);
    }
    __syncthreads();
    if (valid) {
#pragma unroll
      for (int sub = 0; sub < 2; ++sub) {
        const int v0 = st * 32 + sub * 16;
        const float* lrow = lds + (sub * 16 + nloc) * LDS_STRIDE;
        v8f acc0 = {}, acc1 = {};
#pragma unroll
        for (int g = 0; g < 32; g += 2) {
          v2f b0 = *(const v2f*)(lrow + 4 * g + 2 * khalf);
          v2f b1 = *(const v2f*)(lrow + 4 * (g + 1) + 2 * khalf);
          acc0 = __builtin_amdgcn_wmma_f32_16x16x4_f32(
              false, af[g], false, b0, (short)0, acc0, false, false);
          acc1 = __builtin_amdgcn_wmma_f32_16x16x4_f32(
              false, af[g + 1], false, b1, (short)0, acc1, false, false);
        }
        const v8f acc = acc0 + acc1;
        const float cnh = cn[v0 + nloc];
#pragma unroll
        for (int j = 0; j < 8; ++j) {
          const float s = acc[j] - cnh;        // argmax(s) == argmin(dist)
          if (s > best[j]) { best[j] = s; bidx[j] = v0 + nloc; }
        }
      }
    }
  }

  if (valid) {
#pragma unroll
    for (int j = 0; j < 8; ++j) {
      float bv = best[j];
      int bi = bidx[j];
#pragma unroll
      for (int mask = 1; mask <= 8; mask <<= 1) {  // reduce within 16-lane half
        const float ov = __shfl_xor(bv, mask, 32);
        const int   oi = __shfl_xor(bi, mask, 32);
        if (ov > bv || (ov == bv && oi < bi)) { bv = ov; bi = oi; }
      }
      if (nloc == 0) {
        const int mrow = mtile * 16 + j + 8 * khalf;
        tok_f[mrow] = (float)bi;
        tok_i[mrow] = bi;
      }
    }
  }
}

__global__ void k_gather_loss(const int* __restrict__ tok, const float* __restrict__ cb,
                              const float* __restrict__ pe, float* __restrict__ q,
                              float* __restrict__ loss) {
  const int wave = threadIdx.x >> 5, lane = threadIdx.x & 31;
  const int m = blockIdx.x * 8 + wave;
  if (m >= M_TOT) return;
  const int t = tok[m];
  const float4 qv = *(const float4*)(cb + t * D_MODEL + lane * 4);
  const float4 pv = *(const float4*)(pe + m * D_MODEL + lane * 4);
  *(float4*)(q + m * D_MODEL + lane * 4) = qv;
  const float d0 = qv.x - pv.x, d1 = qv.y - pv.y;
  const float d2 = qv.z - pv.z, d3 = qv.w - pv.w;
  float s = d0 * d0 + d1 * d1 + d2 * d2 + d3 * d3;
  for (int mask = 16; mask; mask >>= 1) s += __shfl_xor(s, mask, 32);
  if (lane == 0)
    atomicAdd(loss, s * (2.0f / ((float)M_TOT * (float)D_MODEL)));
}

extern "C" void kernel_launch(void* const* d_in, const int* in_sizes, int n_in,
                              void* d_out, int out_size, void* d_ws, size_t ws_size,
                              hipStream_t stream) {
  const float* x    = (const float*)d_in[0];
  const float* W    = (const float*)d_in[1];
  const float* bias = (const float*)d_in[2];
  const float* cb   = (const float*)d_in[3];

  float* out   = (float*)d_out;
  float* tok_f = out;                               // [B,N] token ids (as float)
  float* pe    = out + M_TOT;                       // [B,N,D] patch_emb
  float* q     = out + M_TOT + M_TOT * D_MODEL;     // [B,N,D] quantized_st
  float* loss  = out + M_TOT + 2 * M_TOT * D_MODEL; // scalar vq_loss

  float* cn  = (float*)d_ws;                          // 8192 floats
  int* tok_i = (int*)((char*)d_ws + VOCAB * sizeof(float));

  k_init      <<<VOCAB / 8, 256, 0, stream>>>(cb, cn, loss);
  k_patch_emb <<<(M_TILES + 7) / 8, 256, 0, stream>>>(x, W, bias, pe);
  k_vq_argmin <<<(M_TILES + 7) / 8, 256, 0, stream>>>(pe, cb, cn, tok_f, tok_i);
  k_gather_loss<<<(M_TOT + 7) / 8, 256, 0, stream>>>(tok_i, cb, pe, q, loss);
}